// SurvivalGNN_32134945309202
// MI455X (gfx1250) — compile-verified
//
#include <hip/hip_runtime.h>
#include <hip/hip_bf16.h>
#include <math.h>

// ---------------- config -----------------
#define NNODES 100000
#define NEDGES 1600000
#define BN_EPS 1e-5f

typedef float v2f __attribute__((ext_vector_type(2)));
typedef float v8f __attribute__((ext_vector_type(8)));

__device__ __forceinline__ float mishf(float x) {
    float sp = (x > 20.0f) ? x : log1pf(expf(x));   // softplus
    return x * tanhf(sp);
}

// ---------------- utility kernels -----------------
__global__ void k_zero(float* __restrict__ p, int n) {
    int i = blockIdx.x * blockDim.x + threadIdx.x;
    if (i < n) p[i] = 0.0f;
}

__global__ void k_degree(const int* __restrict__ dst, float* __restrict__ deg, int e) {
    int i = blockIdx.x * blockDim.x + threadIdx.x;
    if (i < e) atomicAdd(&deg[dst[i]], 1.0f);
}

__global__ void k_rdeg(const float* __restrict__ deg, float* __restrict__ rdeg, int n) {
    int i = blockIdx.x * blockDim.x + threadIdx.x;
    if (i < n) rdeg[i] = 1.0f / fmaxf(deg[i], 1.0f);
}

// Build padded node features h0 [N x 64]:
// cols 0..12 numeric x[:,3:16], 13..28 emb0, 29..44 emb1, 45..60 emb2, 61..63 zero
__global__ void k_embed(const float* __restrict__ x,
                        const float* __restrict__ e0, const float* __restrict__ e1,
                        const float* __restrict__ e2, float* __restrict__ h0, int n) {
    int id = blockIdx.x * blockDim.x + threadIdx.x;
    if (id >= n * 64) return;
    int node = id >> 6;
    int c = id & 63;
    float v = 0.0f;
    if (c < 13) {
        v = x[node * 16 + 3 + c];
    } else if (c < 29) {
        int idx = (int)x[node * 16 + 0]; idx = min(max(idx, 0), 100);
        v = e0[idx * 16 + (c - 13)];
    } else if (c < 45) {
        int idx = (int)x[node * 16 + 1]; idx = min(max(idx, 0), 50);
        v = e1[idx * 16 + (c - 29)];
    } else if (c < 61) {
        int idx = (int)x[node * 16 + 2]; idx = min(max(idx, 0), 20);
        v = e2[idx * 16 + (c - 45)];
    }
    h0[(size_t)node * 64 + c] = v;
}

// Pad [61 x 128] weight to [64 x 128] with zero rows
__global__ void k_padw(const float* __restrict__ w, float* __restrict__ out) {
    int id = blockIdx.x * blockDim.x + threadIdx.x;
    if (id >= 64 * 128) return;
    int k = id >> 7;
    out[id] = (k < 61) ? w[id] : 0.0f;
}

// agg[dst, f..f+3] += h[src, f..f+3] ; float4 gather, lanes coalesced
__global__ void k_scatter4(const float* __restrict__ h, const int* __restrict__ src,
                           const int* __restrict__ dst, float* __restrict__ agg,
                           int total, int logC) {     // logC = log2(F/4)
    int id = blockIdx.x * blockDim.x + threadIdx.x;
    if (id >= total) return;
    int e = id >> logC;
    int c = id & ((1 << logC) - 1);
    int s = src[e];
    int d = dst[e];
    const float4* hp = (const float4*)(h + ((size_t)s << (logC + 2)));
    float4 v = hp[c];
    float* ap = agg + ((size_t)d << (logC + 2)) + c * 4;
    atomicAdd(ap + 0, v.x);
    atomicAdd(ap + 1, v.y);
    atomicAdd(ap + 2, v.z);
    atomicAdd(ap + 3, v.w);
}

// ---------------- fp32 WMMA GEMM -----------------
// out[n x LDO] = (A1 .* rdeg_row) @ B1 [+ A2 @ B2] + bias, optional Mish.
// A row-major (lda = K), B row-major [K x LDO]. One wave owns a 16x64 strip
// (4 accumulator tiles) so each A v2f load feeds 4 (dual: 8) WMMAs; with K/LDO
// as template constants all B addresses fold to immediate offsets.
template<int K, int LDO, bool DUAL>
__global__ __launch_bounds__(256) void k_gemm_wmma(
        const float* __restrict__ A1, const float* __restrict__ rdeg,
        const float* __restrict__ A2,
        const float* __restrict__ B1, const float* __restrict__ B2,
        const float* __restrict__ bias, float* __restrict__ out,
        int n, int domish) {
    int wave = threadIdx.x >> 5;
    int lane = threadIdx.x & 31;
    int rowBase = blockIdx.x * 128 + wave * 16;
    if (rowBase >= n) return;                 // wave-uniform; EXEC stays all-ones
    int colBase = blockIdx.y * 64;
    int half = lane >> 4;                     // 0: holds K0/K1, 1: holds K2/K3
    int m = lane & 15;
    int koff = half * 2;
    int row = rowBase + m;

    float scale = rdeg ? rdeg[row] : 1.0f;
    const float* a1p = A1 + (size_t)row * K + koff;
    const float* a2p = DUAL ? (A2 + (size_t)row * K + koff) : a1p;
    const float* b1p = B1 + (size_t)koff * LDO + colBase + m;
    const float* b2p = DUAL ? (B2 + (size_t)koff * LDO + colBase + m) : b1p;

    __builtin_prefetch(a1p, 0, 0);            // global_prefetch_b8
    if (DUAL) __builtin_prefetch(a2p, 0, 0);

    v8f c0 = {}, c1 = {}, c2 = {}, c3 = {};
    #pragma unroll
    for (int k = 0; k < K; k += 4) {
        v2f a1 = *(const v2f*)(a1p + k);
        a1[0] *= scale; a1[1] *= scale;
        {
            v2f b;
            b[0] = b1p[k * LDO +   0]; b[1] = b1p[k * LDO + LDO +  0];
            c0 = __builtin_amdgcn_wmma_f32_16x16x4_f32(false, a1, false, b, (short)0, c0, false, false);
            b[0] = b1p[k * LDO +  16]; b[1] = b1p[k * LDO + LDO + 16];
            c1 = __builtin_amdgcn_wmma_f32_16x16x4_f32(false, a1, false, b, (short)0, c1, false, false);
            b[0] = b1p[k * LDO +  32]; b[1] = b1p[k * LDO + LDO + 32];
            c2 = __builtin_amdgcn_wmma_f32_16x16x4_f32(false, a1, false, b, (short)0, c2, false, false);
            b[0] = b1p[k * LDO +  48]; b[1] = b1p[k * LDO + LDO + 48];
            c3 = __builtin_amdgcn_wmma_f32_16x16x4_f32(false, a1, false, b, (short)0, c3, false, false);
        }
        if (DUAL) {
            v2f a2 = *(const v2f*)(a2p + k);
            v2f b;
            b[0] = b2p[k * LDO +   0]; b[1] = b2p[k * LDO + LDO +  0];
            c0 = __builtin_amdgcn_wmma_f32_16x16x4_f32(false, a2, false, b, (short)0, c0, false, false);
            b[0] = b2p[k * LDO +  16]; b[1] = b2p[k * LDO + LDO + 16];
            c1 = __builtin_amdgcn_wmma_f32_16x16x4_f32(false, a2, false, b, (short)0, c1, false, false);
            b[0] = b2p[k * LDO +  32]; b[1] = b2p[k * LDO + LDO + 32];
            c2 = __builtin_amdgcn_wmma_f32_16x16x4_f32(false, a2, false, b, (short)0, c2, false, false);
            b[0] = b2p[k * LDO +  48]; b[1] = b2p[k * LDO + LDO + 48];
            c3 = __builtin_amdgcn_wmma_f32_16x16x4_f32(false, a2, false, b, (short)0, c3, false, false);
        }
    }

    // epilogue: bias (+ Mish) and 16-wide coalesced stores per tile column
    v8f acc[4] = {c0, c1, c2, c3};
    #pragma unroll
    for (int g = 0; g < 4; ++g) {
        int col = colBase + g * 16 + m;
        float bv = bias[col];
        #pragma unroll
        for (int i = 0; i < 8; ++i) {
            float v = acc[g][i] + bv;
            if (domish) v = mishf(v);
            out[(size_t)(rowBase + i + half * 8) * LDO + col] = v;
        }
    }
}

// ---------------- BatchNorm (training-mode, biased var) -----------------
__global__ void k_bn_stats(const float* __restrict__ h, float* __restrict__ stats,
                           int n, int F, int chunk) {
    int col = threadIdx.x;                    // blockDim.x == F
    int r0 = blockIdx.x * chunk;
    float s = 0.0f, s2 = 0.0f;
    for (int i = 0; i < chunk; ++i) {
        int r = r0 + i;
        if (r < n) {
            float v = h[(size_t)r * F + col];
            s += v;
            s2 += v * v;
        }
    }
    atomicAdd(&stats[col], s);
    atomicAdd(&stats[F + col], s2);
}

__global__ void k_bn_finalize(const float* __restrict__ stats,
                              const float* __restrict__ g, const float* __restrict__ be,
                              float* __restrict__ ss, int F, float invN) {
    int c = threadIdx.x + blockIdx.x * blockDim.x;
    if (c >= F) return;
    float mu = stats[c] * invN;
    float var = fmaxf(stats[F + c] * invN - mu * mu, 0.0f);
    float sc = g[c] * rsqrtf(var + BN_EPS);
    ss[c] = sc;
    ss[F + c] = be[c] - mu * sc;
}

__global__ void k_bn_apply(const float* __restrict__ in, const float* __restrict__ ss,
                           float* __restrict__ out, int total, int F, int domish) {
    int id = blockIdx.x * blockDim.x + threadIdx.x;
    if (id >= total) return;
    int c = id & (F - 1);
    float v = in[id] * ss[c] + ss[F + c];
    if (domish) v = mishf(v);
    out[id] = v;
}

// ---------------- per-row dot for the 1-wide head outputs -----------------
__global__ void k_rowdot(const float* __restrict__ hid, const float* __restrict__ w,
                         const float* __restrict__ b, float* __restrict__ out,
                         int n, int W) {
    int wave = threadIdx.x >> 5;
    int lane = threadIdx.x & 31;
    int row = blockIdx.x * 8 + wave;
    if (row >= n) return;
    float s = 0.0f;
    for (int j = lane; j < W; j += 32)
        s += hid[(size_t)row * W + j] * w[j];
    #pragma unroll
    for (int o = 16; o > 0; o >>= 1)
        s += __shfl_xor(s, o, 32);
    if (lane == 0) out[row] = s + b[0];
}

// ---------------- orchestration -----------------
static inline int cdiv(int a, int b) { return (a + b - 1) / b; }

extern "C" void kernel_launch(void* const* d_in, const int* in_sizes, int n_in,
                              void* d_out, int out_size, void* d_ws, size_t ws_size,
                              hipStream_t stream) {
    const int N = NNODES, E = NEDGES;
    const float* x   = (const float*)d_in[0];
    const int*   src = (const int*)d_in[1];       // edge_index row 0
    const int*   dst = src + E;                   // edge_index row 1
    const float* e0  = (const float*)d_in[2];
    const float* e1  = (const float*)d_in[3];
    const float* e2  = (const float*)d_in[4];
    const float* Wl0 = (const float*)d_in[5];
    const float* bl0 = (const float*)d_in[6];
    const float* Wr0 = (const float*)d_in[7];
    const float* g0  = (const float*)d_in[8];
    const float* be0 = (const float*)d_in[9];
    const float* Wl1 = (const float*)d_in[10];
    const float* bl1 = (const float*)d_in[11];
    const float* Wr1 = (const float*)d_in[12];
    const float* g1  = (const float*)d_in[13];
    const float* be1 = (const float*)d_in[14];
    const float* Wl2 = (const float*)d_in[15];
    const float* bl2 = (const float*)d_in[16];
    const float* Wr2 = (const float*)d_in[17];
    const float* g2  = (const float*)d_in[18];
    const float* be2 = (const float*)d_in[19];
    const float* pw1 = (const float*)d_in[20];
    const float* pb1 = (const float*)d_in[21];
    const float* pw2 = (const float*)d_in[22];
    const float* pb2 = (const float*)d_in[23];
    const float* aw1 = (const float*)d_in[24];
    const float* ab1 = (const float*)d_in[25];
    const float* aw2 = (const float*)d_in[26];
    const float* ab2 = (const float*)d_in[27];

    float* outp = (float*)d_out;
    float* pred = outp;                         // [N]
    float* emb  = outp + N;                     // [N x 64]
    float* aux  = outp + (size_t)N * 65;        // [N]

    // workspace layout
    float* ws    = (float*)d_ws;
    float* deg   = ws;                          // N
    float* rdeg  = deg  + N;                    // N
    float* h0    = rdeg + N;                    // N*64 (padded input / aux hidden reuse)
    float* hA    = h0   + (size_t)N * 64;       // N*128
    float* hB    = hA   + (size_t)N * 128;      // N*128
    float* agg   = hB   + (size_t)N * 128;      // N*128 (aggregate / predictor hidden reuse)
    float* wpadL = agg  + (size_t)N * 128;      // 64*128
    float* wpadR = wpadL + 64 * 128;            // 64*128
    float* stats = wpadR + 64 * 128;            // 256
    float* ss    = stats + 256;                 // 256

    const int TB = 256;
    const int STAT_CHUNK = 500;                 // 200 blocks cover N
    const float invN = 1.0f / (float)N;
    const int GX = cdiv(N, 128);                // 782 row-blocks

    // degree + reciprocal
    k_zero<<<cdiv(N, TB), TB, 0, stream>>>(deg, N);
    k_degree<<<cdiv(E, TB), TB, 0, stream>>>(dst, deg, E);
    k_rdeg<<<cdiv(N, TB), TB, 0, stream>>>(deg, rdeg, N);

    // padded features + padded layer-0 weights
    k_embed<<<cdiv(N * 64, TB), TB, 0, stream>>>(x, e0, e1, e2, h0, N);
    k_padw<<<cdiv(64 * 128, TB), TB, 0, stream>>>(Wl0, wpadL);
    k_padw<<<cdiv(64 * 128, TB), TB, 0, stream>>>(Wr0, wpadR);

    // ---- layer 0: 64(pad) -> 128 ----
    k_zero<<<cdiv(N * 64, TB), TB, 0, stream>>>(agg, N * 64);
    k_scatter4<<<cdiv(E * 16, TB), TB, 0, stream>>>(h0, src, dst, agg, E * 16, 4);
    k_gemm_wmma<64, 128, true><<<dim3(GX, 2), TB, 0, stream>>>(
        agg, rdeg, h0, wpadL, wpadR, bl0, hA, N, 0);
    k_zero<<<1, 256, 0, stream>>>(stats, 256);
    k_bn_stats<<<cdiv(N, STAT_CHUNK), 128, 0, stream>>>(hA, stats, N, 128, STAT_CHUNK);
    k_bn_finalize<<<1, 128, 0, stream>>>(stats, g0, be0, ss, 128, invN);
    k_bn_apply<<<cdiv(N * 128, TB), TB, 0, stream>>>(hA, ss, hA, N * 128, 128, 1);

    // ---- layer 1: 128 -> 128 ----
    k_zero<<<cdiv(N * 128, TB), TB, 0, stream>>>(agg, N * 128);
    k_scatter4<<<cdiv(E * 32, TB), TB, 0, stream>>>(hA, src, dst, agg, E * 32, 5);
    k_gemm_wmma<128, 128, true><<<dim3(GX, 2), TB, 0, stream>>>(
        agg, rdeg, hA, Wl1, Wr1, bl1, hB, N, 0);
    k_zero<<<1, 256, 0, stream>>>(stats, 256);
    k_bn_stats<<<cdiv(N, STAT_CHUNK), 128, 0, stream>>>(hB, stats, N, 128, STAT_CHUNK);
    k_bn_finalize<<<1, 128, 0, stream>>>(stats, g1, be1, ss, 128, invN);
    k_bn_apply<<<cdiv(N * 128, TB), TB, 0, stream>>>(hB, ss, hB, N * 128, 128, 1);

    // ---- layer 2: 128 -> 64 (node_emb, no Mish) ----
    k_zero<<<cdiv(N * 128, TB), TB, 0, stream>>>(agg, N * 128);
    k_scatter4<<<cdiv(E * 32, TB), TB, 0, stream>>>(hB, src, dst, agg, E * 32, 5);
    k_gemm_wmma<128, 64, true><<<dim3(GX, 1), TB, 0, stream>>>(
        agg, rdeg, hB, Wl2, Wr2, bl2, hA, N, 0);
    k_zero<<<1, 256, 0, stream>>>(stats, 256);
    k_bn_stats<<<cdiv(N, STAT_CHUNK), 64, 0, stream>>>(hA, stats, N, 64, STAT_CHUNK);
    k_bn_finalize<<<1, 64, 0, stream>>>(stats, g2, be2, ss, 64, invN);
    k_bn_apply<<<cdiv(N * 64, TB), TB, 0, stream>>>(hA, ss, emb, N * 64, 64, 0);

    // ---- predictor head: 64 -> 128 (Mish) -> 1 ----
    k_gemm_wmma<64, 128, false><<<dim3(GX, 2), TB, 0, stream>>>(
        emb, nullptr, nullptr, pw1, nullptr, pb1, agg, N, 1);
    k_rowdot<<<cdiv(N, 8), TB, 0, stream>>>(agg, pw2, pb2, pred, N, 128);

    // ---- aux head: 64 -> 64 (Mish) -> 1 ----
    k_gemm_wmma<64, 64, false><<<dim3(GX, 1), TB, 0, stream>>>(
        emb, nullptr, nullptr, aw1, nullptr, ab1, h0, N, 1);
    k_rowdot<<<cdiv(N, 8), TB, 0, stream>>>(h0, aw2, ab2, aux, N, 64);
}